// CausalSelfAttention_76424648065766
// MI455X (gfx1250) — compile-verified
//
#include <hip/hip_runtime.h>
#include <math.h>

// ---------------------------------------------------------------------------
// CDNA5 / gfx1250 causal self-attention, fp32 end-to-end.
// - All matmul FLOPs on V_WMMA_F32_16X16X4_F32 (wave32 matrix pipe).
// - Tile staging via GLOBAL_LOAD_ASYNC_TO_LDS_B128 (ASYNCcnt) with
//   double-buffered LDS so HBM/L2 latency overlaps the WMMA bursts.
// ---------------------------------------------------------------------------

typedef float v2f __attribute__((ext_vector_type(2)));
typedef float v8f __attribute__((ext_vector_type(8)));
typedef int   v4i __attribute__((ext_vector_type(4)));

#define AS1 __attribute__((address_space(1)))
#define AS3 __attribute__((address_space(3)))

#if __has_builtin(__builtin_amdgcn_global_load_async_to_lds_b128) && \
    __has_builtin(__builtin_amdgcn_s_wait_asynccnt)
#define ASYNC_LDS 1
#else
#define ASYNC_LDS 0
#endif

// Async DMA of 16 bytes global -> LDS (per lane). Falls back to a
// synchronous copy if the toolchain lacks the gfx1250 async builtins.
__device__ __forceinline__ void async_b128(const float* gsrc, float* ldst) {
#if ASYNC_LDS
    __builtin_amdgcn_global_load_async_to_lds_b128(
        (AS1 v4i*)(void*)gsrc, (AS3 v4i*)(void*)ldst, 0, 0);
#else
    const float4 t = *(const float4*)gsrc;
    ldst[0] = t.x; ldst[1] = t.y; ldst[2] = t.z; ldst[3] = t.w;
#endif
}

template <int N>
__device__ __forceinline__ void wait_async() {
#if ASYNC_LDS
    __builtin_amdgcn_s_wait_asynccnt(N);
#endif
}

#define WMMA_F32X4(a, b, c) \
    __builtin_amdgcn_wmma_f32_16x16x4_f32(false, (a), false, (b), (short)0, (c), false, false)

__device__ __forceinline__ v8f v8f_zero() {
    v8f v;
#pragma unroll
    for (int i = 0; i < 8; ++i) v[i] = 0.0f;
    return v;
}

// Reductions across a 16-lane half-wave (WMMA C-layout: one row = 16 lanes
// of one half). xor masks 1,2,4,8 stay within a half.
__device__ __forceinline__ float rowmax16(float v) {
#pragma unroll
    for (int m = 8; m >= 1; m >>= 1) v = fmaxf(v, __shfl_xor(v, m, 32));
    return v;
}
__device__ __forceinline__ float rowsum16(float v) {
#pragma unroll
    for (int m = 8; m >= 1; m >>= 1) v += __shfl_xor(v, m, 32);
    return v;
}

// ---------------------------------------------------------------------------
// fp32 WMMA GEMM: C(MxN) = A(MxK, rm) * B(KxN, rm). M,N % 128 == 0, K % 16 == 0.
// 256 threads = 8 waves; block tile 128x128; wave tile 64x32 (4x2 accums).
// Double-buffered async LDS staging; fragments for the whole 16-K stage are
// hoisted ahead of a 32-WMMA burst.
// A-tile stride 20 floats: rows 16B-aligned for b128 async writes, and
// 20*i mod 64 distinct for i<16 -> conflict-free A-fragment reads.
// ---------------------------------------------------------------------------
__launch_bounds__(256, 1)
__global__ void gemm_f32_wmma(const float* __restrict__ A, const float* __restrict__ B,
                              float* __restrict__ C, int M, int N, int K) {
    __shared__ float As[2][128][20];
    __shared__ float Bs[2][16][132];

    const int tid   = threadIdx.x;
    const int lane  = tid & 31;
    const int wave  = tid >> 5;
    const int waveM = wave >> 2;    // 0..1 -> 64-row slab
    const int waveN = wave & 3;     // 0..3 -> 32-col slab
    const int l16   = lane & 15;
    const int lh    = lane >> 4;
    const int mBase = blockIdx.y * 128;
    const int nBase = blockIdx.x * 128;

    // Per-thread staging slots (fixed across stages): 2 b128 for A, 2 for B.
    const int aRow0 = tid >> 2,          aC0 = (tid & 3) << 2;
    const int aRow1 = (tid + 256) >> 2,  aC1 = ((tid + 256) & 3) << 2;
    const int bRow0 = tid >> 5,          bC0 = (tid & 31) << 2;
    const int bRow1 = (tid + 256) >> 5,  bC1 = ((tid + 256) & 31) << 2;

    auto stage = [&](int buf, int kt) {
        async_b128(A + (size_t)(mBase + aRow0) * K + kt + aC0, &As[buf][aRow0][aC0]);
        async_b128(A + (size_t)(mBase + aRow1) * K + kt + aC1, &As[buf][aRow1][aC1]);
        async_b128(B + (size_t)(kt + bRow0) * N + nBase + bC0, &Bs[buf][bRow0][bC0]);
        async_b128(B + (size_t)(kt + bRow1) * N + nBase + bC1, &Bs[buf][bRow1][bC1]);
    };

    v8f acc[4][2];
#pragma unroll
    for (int mi = 0; mi < 4; ++mi)
#pragma unroll
        for (int ni = 0; ni < 2; ++ni) acc[mi][ni] = v8f_zero();

    const int S = K >> 4;
    stage(0, 0);  // prologue

    for (int s = 0; s < S; ++s) {
        const int buf = s & 1;
        if (s + 1 < S) {
            stage(buf ^ 1, (s + 1) << 4);  // stream next tiles during compute
            wait_async<4>();               // the 4 older b128s have landed
        } else {
            wait_async<0>();
        }
        __syncthreads();  // everyone's stage-s tiles visible

        const float (*Asb)[20]  = As[buf];
        const float (*Bsb)[132] = Bs[buf];

        // Hoist all fragments for this 16-K stage, then one WMMA burst.
        v2f af[4][4], bf[4][2];
#pragma unroll
        for (int ks = 0; ks < 4; ++ks) {
            const int k0 = ks * 4 + 2 * lh;  // v0=K(k0), v1=K(k0+1); halves 2 apart
#pragma unroll
            for (int mi = 0; mi < 4; ++mi) {
                const int row = waveM * 64 + mi * 16 + l16;
                af[ks][mi][0] = Asb[row][k0];
                af[ks][mi][1] = Asb[row][k0 + 1];
            }
#pragma unroll
            for (int ni = 0; ni < 2; ++ni) {
                const int col = waveN * 32 + ni * 16 + l16;
                bf[ks][ni][0] = Bsb[k0][col];
                bf[ks][ni][1] = Bsb[k0 + 1][col];
            }
        }
#pragma unroll
        for (int ks = 0; ks < 4; ++ks)
#pragma unroll
            for (int mi = 0; mi < 4; ++mi)
#pragma unroll
                for (int ni = 0; ni < 2; ++ni)
                    acc[mi][ni] = WMMA_F32X4(af[ks][mi], bf[ks][ni], acc[mi][ni]);

        __syncthreads();  // all reads of buf done before it is overwritten
    }

    // C-layout store: row = v + 8*lh, col = l16 within each 16x16 tile.
#pragma unroll
    for (int mi = 0; mi < 4; ++mi)
#pragma unroll
        for (int ni = 0; ni < 2; ++ni) {
            const int col = nBase + waveN * 32 + ni * 16 + l16;
#pragma unroll
            for (int v = 0; v < 8; ++v) {
                const int row = mBase + waveM * 64 + mi * 16 + v + 8 * lh;
                C[(size_t)row * N + col] = acc[mi][ni][v];
            }
        }
}

// ---------------------------------------------------------------------------
// RoPE + head split. qkv (B,T,3C) -> Q,K (rotated), V in (B,H,T,D) layout.
// ---------------------------------------------------------------------------
__global__ void rope_split_kernel(const float* __restrict__ qkv, float* __restrict__ Q,
                                  float* __restrict__ Ko, float* __restrict__ Vo) {
    const int idx = blockIdx.x * 256 + threadIdx.x;  // B*T*H*64 total
    const int d2 = idx & 63;
    const int h  = (idx >> 6) & 15;
    const int t  = (idx >> 10) & 2047;
    const int b  = idx >> 21;

    const float inv_freq = __expf(-(float)d2 * (9.210340371976184f / 64.0f));
    const float ang = (float)t * inv_freq;
    float s, c;
    __sincosf(ang, &s, &c);

    const size_t row = ((size_t)b * 2048 + t) * 6144;  // 3C
    const int col = h * 128 + d2;
    const float q1 = qkv[row + col],        q2 = qkv[row + col + 64];
    const float k1 = qkv[row + 2048 + col], k2 = qkv[row + 2048 + col + 64];
    const float v1 = qkv[row + 4096 + col], v2 = qkv[row + 4096 + col + 64];

    const size_t o = (((size_t)b * 16 + h) * 2048 + t) * 128 + d2;
    Q[o]       = q1 * c - q2 * s;
    Q[o + 64]  = q2 * c + q1 * s;
    Ko[o]      = k1 * c - k2 * s;
    Ko[o + 64] = k2 * c + k1 * s;
    Vo[o]      = v1;
    Vo[o + 64] = v2;
}

// ---------------------------------------------------------------------------
// Flash attention (causal), fp32 WMMA. 4 waves/block; each wave owns a
// 16-row Q tile (Q A-fragments resident in 64 VGPRs). 32-row KV chunks,
// double-buffered async LDS staging. Online softmax via half-wave shuffles
// on the WMMA C-layout. Causal branch is wave-uniform -> EXEC all-1s.
// ---------------------------------------------------------------------------
__launch_bounds__(128, 1)
__global__ void flash_attn_kernel(const float* __restrict__ Qg, const float* __restrict__ Kg,
                                  const float* __restrict__ Vg, float* __restrict__ Y) {
    constexpr int T = 2048, D = 128, Cfull = 2048;
    __shared__ float Ks[2][32][132];
    __shared__ float Vs[2][32][132];
    __shared__ float Ps[4][16][33];  // per-wave P transpose scratch

    const int tid  = threadIdx.x;
    const int lane = tid & 31;
    const int wave = tid >> 5;      // 0..3
    const int l16  = lane & 15;
    const int lh   = lane >> 4;
    const int bh   = blockIdx.y;    // b*16 + h
    const int qi0  = blockIdx.x * 64 + wave * 16;

    const float* Qp = Qg + (size_t)bh * T * D;
    const float* Kp = Kg + (size_t)bh * T * D;
    const float* Vp = Vg + (size_t)bh * T * D;

    auto stageKV = [&](int buf, int j0) {
#pragma unroll
        for (int r = 0; r < 8; ++r) {
            const int id = tid + r * 128;
            const int row = id >> 5, c0 = (id & 31) << 2;
            async_b128(Kp + (size_t)(j0 + row) * D + c0, &Ks[buf][row][c0]);
            async_b128(Vp + (size_t)(j0 + row) * D + c0, &Vs[buf][row][c0]);
        }
    };

    // Preload 16x128 Q tile as 32 A-fragments (K=4 each).
    v2f qf[32];
    {
        const float* qrow = Qp + (size_t)(qi0 + l16) * D + 2 * lh;
#pragma unroll
        for (int kb = 0; kb < 32; ++kb) {
            const float2 t2 = *(const float2*)(qrow + kb * 4);
            qf[kb][0] = t2.x;
            qf[kb][1] = t2.y;
        }
    }

    v8f o[8];
#pragma unroll
    for (int dt = 0; dt < 8; ++dt) o[dt] = v8f_zero();
    float mrow[8], lrow[8];
#pragma unroll
    for (int v = 0; v < 8; ++v) { mrow[v] = -INFINITY; lrow[v] = 0.0f; }

    const float scale = 0.08838834764831845f;  // 1/sqrt(128)
    const int jEnd = blockIdx.x * 64 + 63;     // last q row served by this block

    stageKV(0, 0);  // prologue

    for (int j0 = 0; j0 <= jEnd; j0 += 32) {
        const int buf = (j0 >> 5) & 1;
        if (j0 + 32 <= jEnd) {
            stageKV(buf ^ 1, j0 + 32);  // stream next chunk during compute
            wait_async<16>();           // current chunk's 16 b128s have landed
        } else {
            wait_async<0>();
        }
        __syncthreads();

        const float (*Kb)[132] = Ks[buf];
        const float (*Vb)[132] = Vs[buf];

        if (j0 <= qi0 + 15) {  // wave-uniform
            // S = Q * K^T for two 16-col tiles (64 WMMAs over K=128).
            v8f s0 = v8f_zero(), s1 = v8f_zero();
#pragma unroll
            for (int kb = 0; kb < 32; ++kb) {
                const int k0 = kb * 4 + 2 * lh;
                v2f b0, b1;
                b0[0] = Kb[l16][k0];      b0[1] = Kb[l16][k0 + 1];
                b1[0] = Kb[16 + l16][k0]; b1[1] = Kb[16 + l16][k0 + 1];
                s0 = WMMA_F32X4(qf[kb], b0, s0);
                s1 = WMMA_F32X4(qf[kb], b1, s1);
            }

            // Causal mask + online softmax; C-layout row = v + 8*lh, col = l16.
            float sfac[8];
#pragma unroll
            for (int v = 0; v < 8; ++v) {
                const int qrow = qi0 + v + 8 * lh;
                float a0 = s0[v] * scale;
                float a1 = s1[v] * scale;
                if (j0 + l16 > qrow)      a0 = -INFINITY;
                if (j0 + 16 + l16 > qrow) a1 = -INFINITY;
                const float rm = rowmax16(fmaxf(a0, a1));
                const float mo = mrow[v];
                const float mn = fmaxf(mo, rm);
                const float sc = __expf(mo - mn);
                const float p0 = __expf(a0 - mn);
                const float p1 = __expf(a1 - mn);
                lrow[v] = lrow[v] * sc + rowsum16(p0 + p1);
                mrow[v] = mn;
                sfac[v] = sc;
                Ps[wave][v + 8 * lh][l16]      = p0;  // C-layout -> row-major
                Ps[wave][v + 8 * lh][16 + l16] = p1;
            }

            // Rescale running output, then O += P * V (64 WMMAs over K=32).
#pragma unroll
            for (int dt = 0; dt < 8; ++dt)
#pragma unroll
                for (int v = 0; v < 8; ++v) o[dt][v] *= sfac[v];

#pragma unroll
            for (int kb = 0; kb < 8; ++kb) {
                const int k0 = kb * 4 + 2 * lh;
                v2f a;
                a[0] = Ps[wave][l16][k0];
                a[1] = Ps[wave][l16][k0 + 1];
#pragma unroll
                for (int dt = 0; dt < 8; ++dt) {
                    v2f b;
                    b[0] = Vb[k0][dt * 16 + l16];
                    b[1] = Vb[k0 + 1][dt * 16 + l16];
                    o[dt] = WMMA_F32X4(a, b, o[dt]);
                }
            }
        }
        __syncthreads();  // all reads of buf done before it is overwritten
    }

    // Epilogue: normalize and store to (B,T,C) with c = h*128 + d.
    const int b = bh >> 4, h = bh & 15;
#pragma unroll
    for (int v = 0; v < 8; ++v) {
        const float inv = 1.0f / lrow[v];
        const int trow = qi0 + v + 8 * lh;
        float* yrow = Y + ((size_t)b * T + trow) * Cfull + h * D;
#pragma unroll
        for (int dt = 0; dt < 8; ++dt) yrow[dt * 16 + l16] = o[dt][v] * inv;
    }
}

// ---------------------------------------------------------------------------
// Host-side launcher.
// ---------------------------------------------------------------------------
extern "C" void kernel_launch(void* const* d_in, const int* in_sizes, int n_in,
                              void* d_out, int out_size, void* d_ws, size_t ws_size,
                              hipStream_t stream) {
    (void)in_sizes; (void)n_in; (void)out_size; (void)ws_size;
    constexpr int B = 2, T = 2048, C = 2048;
    constexpr int M = B * T;  // 4096

    const float* x     = (const float*)d_in[0];
    const float* Wqkv  = (const float*)d_in[1];
    const float* Wproj = (const float*)d_in[2];
    float* out = (float*)d_out;

    float* ws    = (float*)d_ws;
    float* qkv   = ws;                       // B*T*3C floats
    float* Qh    = ws + (size_t)M * 3 * C;   // B*T*C each
    float* Kh    = Qh + (size_t)M * C;
    float* Vh    = Kh + (size_t)M * C;
    float* Yattn = ws;                       // reuse qkv region (dead after RoPE)

    // 1) qkv = x @ Wqkv : (4096 x 2048) * (2048 x 6144)
    gemm_f32_wmma<<<dim3(3 * C / 128, M / 128), 256, 0, stream>>>(x, Wqkv, qkv, M, 3 * C, C);

    // 2) RoPE + head split into (B,H,T,D)
    rope_split_kernel<<<(B * T * 16 * 64) / 256, 256, 0, stream>>>(qkv, Qh, Kh, Vh);

    // 3) Flash attention -> Yattn in (B,T,C)
    flash_attn_kernel<<<dim3(T / 64, B * 16), 128, 0, stream>>>(Qh, Kh, Vh, Yattn);

    // 4) out = Yattn @ Wproj : (4096 x 2048) * (2048 x 2048)
    gemm_f32_wmma<<<dim3(C / 128, M / 128), 256, 0, stream>>>(Yattn, Wproj, out, M, C, C);
}